// TPD_19112604467812
// MI455X (gfx1250) — compile-verified
//
#include <hip/hip_runtime.h>
#include <hip/hip_bf16.h>
#include <math.h>

typedef __attribute__((ext_vector_type(2))) float v2f;
typedef __attribute__((ext_vector_type(8))) float v8f;
typedef unsigned int u32x4 __attribute__((ext_vector_type(4)));
typedef int          i32x8 __attribute__((ext_vector_type(8)));
typedef int          i32x4 __attribute__((ext_vector_type(4)));

#define SINK_ALPHA   20.0f
#define INV_ALPHA    (1.0f / 20.0f)
#define EPS_R        1e-16f
#define STOP_THR     0.005f
#define MAX_ITER     200

#define DIMK   384     // embedding dim (fixed by the reference)
#define KC     32      // k-chunk staged per TDM tile
#define RS     36      // padded LDS row stride in dwords (32 data + 4 TDM pad)
#define TILE_DW (64 * RS)   // dwords per 64-row tile in LDS

// ---------------------------------------------------------------------------
// Tensor Data Mover: DMA one 64-row x 32-col fp32 tile (row-major tensor with
// row stride DIMK) into LDS at byte offset lds_byte. The descriptor enables
// LDS padding (4 dwords after every 32) so LDS rows have stride 36 dwords:
// gcd(36,64)=4 keeps the 16-lane b64 fragment reads bank-conflict-free and
// every fragment address 8B-aligned. Tracked by TENSORcnt.
// ---------------------------------------------------------------------------
__device__ __forceinline__ void tdm_load_tile32(unsigned lds_byte, const float* g, int rows) {
    unsigned long long ga = (unsigned long long)(uintptr_t)g;
    u32x4 g0;
    g0[0] = 1u;                                            // count=1, user mode
    g0[1] = lds_byte;                                      // lds_addr
    g0[2] = (unsigned)ga;                                  // global_addr[31:0]
    g0[3] = (unsigned)((ga >> 32) & 0x01FFFFFFu) | (2u << 30);  // addr[56:32] | type=2
    i32x8 g1;
    g1[0] = (2 << 16)      // data_size = 4B
          | (1 << 20)      // pad_enable
          | (4 << 22)      // pad_interval code 4 -> every 32 dwords
          | (3 << 25);     // pad_amount  code 3 -> 4 dwords
    g1[1] = (int)((DIMK & 0xFFFF) << 16);                  // tensor_dim0[15:0] (bits 79:48)
    g1[2] = (int)(((unsigned)rows & 0xFFFFu) << 16);       // dim0 hi=0 | tensor_dim1[15:0]
    g1[3] = (int)((KC & 0xFFFF) << 16) | ((rows >> 16) & 0xFFFF); // dim1 hi | tile_dim0
    g1[4] = 64;                                            // tile_dim1=64, tile_dim2=0
    g1[5] = DIMK;                                          // tensor_dim0_stride lo
    g1[6] = 0;                                             // stride hi | dim1_stride lo
    g1[7] = 0;
    i32x4 gz = {0, 0, 0, 0};                               // 2D tensor: groups 2/3 unused
#if defined(__clang_major__) && (__clang_major__ >= 23)
    i32x8 g4 = {0, 0, 0, 0, 0, 0, 0, 0};
    __builtin_amdgcn_tensor_load_to_lds(g0, g1, gz, gz, g4, 0);
#else
    __builtin_amdgcn_tensor_load_to_lds(g0, g1, gz, gz, 0);
#endif
}

// ---------------------------------------------------------------------------
// Row squared-norms: one block per row, LDS tree reduction (fixed order).
// ---------------------------------------------------------------------------
__global__ void tpd_norms_kernel(const float* __restrict__ X, float* __restrict__ out, int d) {
    __shared__ float red[256];
    const float* row = X + (size_t)blockIdx.x * d;
    float s = 0.f;
    for (int j = threadIdx.x; j < d; j += 256) { float x = row[j]; s += x * x; }
    red[threadIdx.x] = s; __syncthreads();
    for (int w = 128; w > 0; w >>= 1) {
        if (threadIdx.x < w) red[threadIdx.x] += red[threadIdx.x + w];
        __syncthreads();
    }
    if (threadIdx.x == 0) out[blockIdx.x] = red[0];
}

// ---------------------------------------------------------------------------
// K = exp(-alpha * (||x_i||^2 + ||y_j||^2 - 2 x_i.y_j))
// One wave per 64x64 tile. Double-buffered TDM stages X/Y 64x32 tiles into
// padded LDS; fragments come from ds_load_b64; math is chained
// V_WMMA_F32_16X16X4_F32 (full fp32 — M feeds exp(-20*M), so reduced
// precision would blow up as e^{20*dM}).
// A frag (16x4 f32): lane L holds row (L&15), K pair (k + 2*(L>>4)) -> v2f.
// C/D (16x16 f32): v8f, element r -> row r + 8*(L>>4), col (L&15).
// ---------------------------------------------------------------------------
__global__ __launch_bounds__(32)
void tpd_gemm_exp_kernel(const float* __restrict__ X, const float* __restrict__ Y,
                         const float* __restrict__ xs, const float* __restrict__ ys,
                         float* __restrict__ Kout, int n, int m) {
    __shared__ float lds[4 * TILE_DW];    // X0 | Y0 | X1 | Y1 : 36 KB
    const int lane = threadIdx.x;
    const int half = lane >> 4;           // 0 or 1
    const int l16  = lane & 15;
    const int r0 = blockIdx.y * 64;
    const int c0 = blockIdx.x * 64;

    const unsigned lds_base = (unsigned)(uintptr_t)&lds[0];

    // prime buffer 0
    tdm_load_tile32(lds_base,                        X + (size_t)r0 * DIMK, n);
    tdm_load_tile32(lds_base + TILE_DW * 4u,         Y + (size_t)c0 * DIMK, m);

    v8f acc[4][4] = {};
    const int nchunk = DIMK / KC;   // 12
    for (int c = 0; c < nchunk; ++c) {
        const int tb = (c & 1) ? 2 * TILE_DW : 0;   // tile base (dwords) being consumed
        if (c + 1 < nchunk) {
            const int pb = (c & 1) ? 0 : 2 * TILE_DW;   // buffer being refilled
            // make sure in-flight LDS reads of that buffer have drained
            asm volatile("s_wait_dscnt 0x0" ::: "memory");
            tdm_load_tile32(lds_base + (unsigned)pb * 4u,
                            X + (size_t)r0 * DIMK + (c + 1) * KC, n);
            tdm_load_tile32(lds_base + (unsigned)(pb + TILE_DW) * 4u,
                            Y + (size_t)c0 * DIMK + (c + 1) * KC, m);
            __builtin_amdgcn_s_wait_tensorcnt(2);   // current chunk's 2 tiles are done
        } else {
            __builtin_amdgcn_s_wait_tensorcnt(0);
        }
#pragma unroll
        for (int ks = 0; ks < KC / 4; ++ks) {
            v2f af[4], bf[4];
#pragma unroll
            for (int a = 0; a < 4; ++a)
                af[a] = *(const v2f*)&lds[tb + (a * 16 + l16) * RS + ks * 4 + half * 2];
#pragma unroll
            for (int b = 0; b < 4; ++b)
                bf[b] = *(const v2f*)&lds[tb + TILE_DW + (b * 16 + l16) * RS + ks * 4 + half * 2];
#pragma unroll
            for (int a = 0; a < 4; ++a)
#pragma unroll
                for (int b = 0; b < 4; ++b)
                    acc[a][b] = __builtin_amdgcn_wmma_f32_16x16x4_f32(
                        /*neg_a=*/false, af[a], /*neg_b=*/false, bf[b],
                        /*c_mod=*/(short)0, acc[a][b], /*reuse=*/false, false);
        }
    }

#pragma unroll
    for (int a = 0; a < 4; ++a) {
#pragma unroll
        for (int b = 0; b < 4; ++b) {
            const int col = c0 + b * 16 + l16;
            const float yn = ys[col];
#pragma unroll
            for (int r = 0; r < 8; ++r) {
                const int row = r0 + a * 16 + r + half * 8;
                const float Mij = xs[row] + yn - 2.0f * acc[a][b][r];
                Kout[(size_t)row * m + col] = expf(-SINK_ALPHA * Mij);
            }
        }
    }
}

// ---------------------------------------------------------------------------
// Sinkhorn init: u = 1/n, v = 0, active = 1.
// ---------------------------------------------------------------------------
__global__ void tpd_init_kernel(float* __restrict__ u, int n, float uval,
                                float* __restrict__ v, int m, int* __restrict__ active) {
    const int idx = blockIdx.x * blockDim.x + threadIdx.x;
    if (idx < n) u[idx] = uval;
    if (idx < m) v[idx] = 0.f;
    if (idx == 0) *active = 1;
}

// ---------------------------------------------------------------------------
// Fused pair0+pair1: v_j = (1/m) / ((K^T u)_j + eps). Thread-per-column,
// coalesced; grid split [0,blocks0) -> pair0, rest -> pair1.
// ---------------------------------------------------------------------------
__global__ void tpd_colv2_kernel(const float* __restrict__ Ka, const float* __restrict__ ua,
                                 float* __restrict__ va, int na, int ma, float ba,
                                 const int* __restrict__ acta,
                                 const float* __restrict__ Kb, const float* __restrict__ ub,
                                 float* __restrict__ vb, int nb, int mb, float bb,
                                 const int* __restrict__ actb, int blocks0) {
    const float* Km; const float* u; float* v; int n, m, j; float binv; const int* act;
    if ((int)blockIdx.x < blocks0) {
        Km = Ka; u = ua; v = va; n = na; m = ma; binv = ba; act = acta;
        j = blockIdx.x * 256 + threadIdx.x;
    } else {
        Km = Kb; u = ub; v = vb; n = nb; m = mb; binv = bb; act = actb;
        j = (blockIdx.x - blocks0) * 256 + threadIdx.x;
    }
    if (*act == 0) return;
    const float* p = Km + j;
    float t = 0.f;
    for (int i = 0; i < n; ++i) { t += (*p) * u[i]; p += m; }
    v[j] = binv / (t + EPS_R);
}

// ---------------------------------------------------------------------------
// Fused pair0+pair1: u_i = (1/n) / ((K v)_i + eps). Block-per-row, LDS tree.
// ---------------------------------------------------------------------------
__global__ void tpd_rowu2_kernel(const float* __restrict__ Ka, const float* __restrict__ va,
                                 float* __restrict__ ua, int na, int ma, float aa,
                                 const int* __restrict__ acta,
                                 const float* __restrict__ Kb, const float* __restrict__ vb,
                                 float* __restrict__ ub, int mb, float ab,
                                 const int* __restrict__ actb) {
    const float* Km; const float* v; float* u; int m, row; float ainv; const int* act;
    if ((int)blockIdx.x < na) {
        Km = Ka; v = va; u = ua; m = ma; ainv = aa; act = acta; row = blockIdx.x;
    } else {
        Km = Kb; v = vb; u = ub; m = mb; ainv = ab; act = actb; row = blockIdx.x - na;
    }
    if (*act == 0) return;
    __shared__ float red[256];
    const float* p = Km + (size_t)row * m;
    float s = 0.f;
    for (int j = threadIdx.x; j < m; j += 256) s += p[j] * v[j];
    red[threadIdx.x] = s; __syncthreads();
    for (int w = 128; w > 0; w >>= 1) {
        if (threadIdx.x < w) red[threadIdx.x] += red[threadIdx.x + w];
        __syncthreads();
    }
    if (threadIdx.x == 0) u[row] = ainv / (red[0] + EPS_R);
}

// ---------------------------------------------------------------------------
// Fused err partials: e_j = |v_j * (K^T u)_j - 1/m| summed per block.
// Pair0 partials at [0,blocks0), pair1 at [blocks0, ...).
// ---------------------------------------------------------------------------
__global__ void tpd_errp2_kernel(const float* __restrict__ Ka, const float* __restrict__ ua,
                                 const float* __restrict__ va, int na, int ma, float ba,
                                 const int* __restrict__ acta,
                                 const float* __restrict__ Kb, const float* __restrict__ ub,
                                 const float* __restrict__ vb, int nb, int mb, float bb,
                                 const int* __restrict__ actb,
                                 float* __restrict__ partials, int blocks0) {
    const float* Km; const float* u; const float* v; int n, m, j; float bval; const int* act;
    if ((int)blockIdx.x < blocks0) {
        Km = Ka; u = ua; v = va; n = na; m = ma; bval = ba; act = acta;
        j = blockIdx.x * 256 + threadIdx.x;
    } else {
        Km = Kb; u = ub; v = vb; n = nb; m = mb; bval = bb; act = actb;
        j = (blockIdx.x - blocks0) * 256 + threadIdx.x;
    }
    if (*act == 0) return;
    __shared__ float red[256];
    const float* p = Km + j;
    float t = 0.f;
    for (int i = 0; i < n; ++i) { t += (*p) * u[i]; p += m; }
    red[threadIdx.x] = fabsf(v[j] * t - bval); __syncthreads();
    for (int w = 128; w > 0; w >>= 1) {
        if (threadIdx.x < w) red[threadIdx.x] += red[threadIdx.x + w];
        __syncthreads();
    }
    if (threadIdx.x == 0) partials[blockIdx.x] = red[0];
}

__global__ void tpd_errred2_kernel(const float* __restrict__ partials, int blocks0, int blocks1,
                                   int* __restrict__ act0, int* __restrict__ act1) {
    if (threadIdx.x == 0) {
        if (*act0) {
            float e = 0.f;
            for (int i = 0; i < blocks0; ++i) e += partials[i];
            if (e <= STOP_THR) *act0 = 0;   // freeze: matches reference masked updates
        }
    } else if (threadIdx.x == 1) {
        if (*act1) {
            float e = 0.f;
            for (int i = 0; i < blocks1; ++i) e += partials[blocks0 + i];
            if (e <= STOP_THR) *act1 = 0;
        }
    }
}

// ---------------------------------------------------------------------------
// transp_ij = u_i K_ij v_j  (pair0: store only)
// ---------------------------------------------------------------------------
__global__ void tpd_transp_kernel(const float* __restrict__ Kmat, const float* __restrict__ u,
                                  const float* __restrict__ v, float* __restrict__ out,
                                  int n, int m) {
    const size_t idx = (size_t)blockIdx.x * blockDim.x + threadIdx.x;
    if (idx >= (size_t)n * m) return;
    const int i = (int)(idx / m);
    const int j = (int)(idx - (size_t)i * m);
    out[idx] = u[i] * Kmat[idx] * v[j];
}

// ---------------------------------------------------------------------------
// Pair1: write transp and accumulate loss = sum transp * M, with
// M = -log(K)/alpha where K > 0 (K == 0 => transp == 0 => zero contribution).
// Fixed grid + fixed-order block partials -> bit-deterministic.
// ---------------------------------------------------------------------------
__global__ void tpd_transp_loss_kernel(const float* __restrict__ Kmat, const float* __restrict__ u,
                                       const float* __restrict__ v, float* __restrict__ out,
                                       float* __restrict__ partials, int n, int m) {
    __shared__ float red[256];
    const size_t total  = (size_t)n * m;
    const size_t stride = (size_t)gridDim.x * blockDim.x;
    float acc = 0.f;
    for (size_t idx = (size_t)blockIdx.x * blockDim.x + threadIdx.x; idx < total; idx += stride) {
        const int i = (int)(idx / m);
        const int j = (int)(idx - (size_t)i * m);
        const float k = Kmat[idx];
        const float t = u[i] * k * v[j];
        out[idx] = t;
        if (k > 0.f) acc += t * (-logf(k) * INV_ALPHA);
    }
    red[threadIdx.x] = acc; __syncthreads();
    for (int w = 128; w > 0; w >>= 1) {
        if (threadIdx.x < w) red[threadIdx.x] += red[threadIdx.x + w];
        __syncthreads();
    }
    if (threadIdx.x == 0) partials[blockIdx.x] = red[0];
}

__global__ void tpd_lossred_kernel(const float* __restrict__ partials, int nblocks,
                                   float* __restrict__ out_loss) {
    __shared__ float red[256];
    float s = 0.f;
    for (int i = threadIdx.x; i < nblocks; i += 256) s += partials[i];
    red[threadIdx.x] = s; __syncthreads();
    for (int w = 128; w > 0; w >>= 1) {
        if (threadIdx.x < w) red[threadIdx.x] += red[threadIdx.x + w];
        __syncthreads();
    }
    if (threadIdx.x == 0) out_loss[0] = red[0];
}

// ---------------------------------------------------------------------------
extern "C" void kernel_launch(void* const* d_in, const int* in_sizes, int n_in,
                              void* d_out, int out_size, void* d_ws, size_t ws_size,
                              hipStream_t stream) {
    (void)in_sizes; (void)n_in; (void)out_size; (void)ws_size;
    const float* e0 = (const float*)d_in[0];   // 512  x 384
    const float* e1 = (const float*)d_in[1];   // 2048 x 384
    const float* e2 = (const float*)d_in[2];   // 8192 x 384
    const int N0 = 512, N1 = 2048, N2 = 8192, D = 384;

    // Workspace layout (floats)
    float* ws = (float*)d_ws;
    size_t off = 0;
    float* K0   = ws + off; off += (size_t)N0 * N1;       // 4 MB
    float* K1   = ws + off; off += (size_t)N1 * N2;       // 64 MB (L2 resident: 192MB L2)
    float* nrm0 = ws + off; off += N0;
    float* nrm1 = ws + off; off += N1;
    float* nrm2 = ws + off; off += N2;
    float* u0   = ws + off; off += N0;
    float* v0   = ws + off; off += N1;
    float* u1   = ws + off; off += N1;
    float* v1   = ws + off; off += N2;
    float* errp = ws + off; off += 64;                    // err block partials (40 used)
    float* lossp= ws + off; off += 2048;                  // loss block partials
    int*   act0 = (int*)(ws + off); off += 1;
    int*   act1 = (int*)(ws + off); off += 1;

    float* out_loss = (float*)d_out;
    float* out_t0   = out_loss + 1;
    float* out_t1   = out_t0 + (size_t)N0 * N1;

    // 1) row norms
    tpd_norms_kernel<<<N0, 256, 0, stream>>>(e0, nrm0, D);
    tpd_norms_kernel<<<N1, 256, 0, stream>>>(e1, nrm1, D);
    tpd_norms_kernel<<<N2, 256, 0, stream>>>(e2, nrm2, D);

    // 2) K matrices: TDM-staged fp32 WMMA GEMM + fused exp epilogue
    {
        dim3 g0(N1 / 64, N0 / 64);
        tpd_gemm_exp_kernel<<<g0, 32, 0, stream>>>(e0, e1, nrm0, nrm1, K0, N0, N1);
        dim3 g1(N2 / 64, N1 / 64);
        tpd_gemm_exp_kernel<<<g1, 32, 0, stream>>>(e1, e2, nrm1, nrm2, K1, N1, N2);
    }

    // 3) Sinkhorn init
    tpd_init_kernel<<<(N1 + 255) / 256, 256, 0, stream>>>(u0, N0, 1.0f / N0, v0, N1, act0);
    tpd_init_kernel<<<(N2 + 255) / 256, 256, 0, stream>>>(u1, N1, 1.0f / N1, v1, N2, act1);

    // 4) 200 fixed iterations, both pairs fused per launch, device-side freeze
    const float b0 = 1.0f / N1, a0 = 1.0f / N0;
    const float b1 = 1.0f / N2, a1 = 1.0f / N1;
    const int cb0 = N1 / 256, cb1 = N2 / 256;   // 8 + 32 col-blocks
    for (int it = 1; it <= MAX_ITER; ++it) {
        tpd_colv2_kernel<<<cb0 + cb1, 256, 0, stream>>>(
            K0, u0, v0, N0, N1, b0, act0,
            K1, u1, v1, N1, N2, b1, act1, cb0);
        tpd_rowu2_kernel<<<N0 + N1, 256, 0, stream>>>(
            K0, v0, u0, N0, N1, a0, act0,
            K1, v1, u1, N2, a1, act1);
        if (it % 50 == 1) {   // reference: err refreshed when cpt_new % 50 == 1
            tpd_errp2_kernel<<<cb0 + cb1, 256, 0, stream>>>(
                K0, u0, v0, N0, N1, b0, act0,
                K1, u1, v1, N1, N2, b1, act1, errp, cb0);
            tpd_errred2_kernel<<<1, 2, 0, stream>>>(errp, cb0, cb1, act0, act1);
        }
    }

    // 5) transport plans + final-pair loss
    {
        const size_t tot0 = (size_t)N0 * N1;
        tpd_transp_kernel<<<(unsigned)((tot0 + 255) / 256), 256, 0, stream>>>(K0, u0, v0, out_t0, N0, N1);
        tpd_transp_loss_kernel<<<2048, 256, 0, stream>>>(K1, u1, v1, out_t1, lossp, N1, N2);
        tpd_lossred_kernel<<<1, 256, 0, stream>>>(lossp, 2048, out_loss);
    }
}